// MobileNetV2_50405736186389
// MI455X (gfx1250) — compile-verified
//
#include <hip/hip_runtime.h>
#include <hip/hip_bf16.h>

typedef float v2f __attribute__((ext_vector_type(2)));
typedef float v8f __attribute__((ext_vector_type(8)));

#define B_    4
#define C_    64
#define R_    4
#define T_    128
#define HW_   196      // 14*14
#define HWP_  208      // padded to 13 tiles of 16
#define Q_    25088    // T_*HW_

// ---------------------------------------------------------------------------
// K1: pointwise channel GEMMs via fp32 WMMA.
//   x2[b,o,q] = sum_c W_down2[o,c] * x[b,c,q]    (64x64 x Q)
//   xr[b,r,q] = sum_c W_down [r,c] * x[b,c,q]    (4x64  x Q, padded to 16 rows)
// Block = 256 threads (8 waves), handles 128 positions of one batch b.
// ---------------------------------------------------------------------------
__global__ void k1_pointwise(const float* __restrict__ x,
                             const float* __restrict__ Wd2,
                             const float* __restrict__ Wd,
                             float* __restrict__ x2,
                             float* __restrict__ xr)
{
    __shared__ float Wd2s[64 * 64];   // [o][c]
    __shared__ float Wdps[16 * 64];   // W_down padded to 16 rows
    __shared__ float Xs[64 * 128];    // [c][pos]

    const int tid   = threadIdx.x;
    const int b     = blockIdx.y;
    const int qbase = blockIdx.x * 128;

    for (int i = tid; i < 64 * 64; i += 256) Wd2s[i] = Wd2[i];
    for (int i = tid; i < 16 * 64; i += 256) Wdps[i] = (i < 4 * 64) ? Wd[i] : 0.0f;
    for (int i = tid; i < 64 * 128; i += 256) {
        int c = i >> 7, col = i & 127;
        Xs[i] = x[(size_t)b * (64 * Q_) + (size_t)c * Q_ + qbase + col];
    }
    __syncthreads();

    const int lane  = tid & 31;
    const int pt    = tid >> 5;        // wave id == position-tile id (0..7)
    const int row   = lane & 15;       // M for A-op, N for B-op / C-D
    const int hi    = lane >> 4;       // 0: K{0,1} / M=v ; 1: K{2,3} / M=v+8
    const int khalf = hi * 2;
    const int hi8   = hi * 8;
    const int q     = qbase + pt * 16 + row;

    // ---- x2: 4 output-channel tiles of 16 ----
    for (int mc = 0; mc < 4; ++mc) {
        v8f acc = {};
#pragma unroll
        for (int k4 = 0; k4 < 16; ++k4) {
            const int kk = k4 * 4 + khalf;
            v2f a, bb;
            a.x  = Wd2s[(mc * 16 + row) * 64 + kk];
            a.y  = Wd2s[(mc * 16 + row) * 64 + kk + 1];
            bb.x = Xs[kk * 128 + pt * 16 + row];
            bb.y = Xs[(kk + 1) * 128 + pt * 16 + row];
            acc = __builtin_amdgcn_wmma_f32_16x16x4_f32(false, a, false, bb,
                                                        (short)0, acc, false, false);
        }
#pragma unroll
        for (int v = 0; v < 8; ++v) {
            const int c = mc * 16 + v + hi8;
            x2[(size_t)b * (64 * Q_) + (size_t)c * Q_ + q] = acc[v];
        }
    }

    // ---- xr: one padded 16-row tile (rows 0..3 valid) ----
    {
        v8f acc = {};
#pragma unroll
        for (int k4 = 0; k4 < 16; ++k4) {
            const int kk = k4 * 4 + khalf;
            v2f a, bb;
            a.x  = Wdps[row * 64 + kk];
            a.y  = Wdps[row * 64 + kk + 1];
            bb.x = Xs[kk * 128 + pt * 16 + row];
            bb.y = Xs[(kk + 1) * 128 + pt * 16 + row];
            acc = __builtin_amdgcn_wmma_f32_16x16x4_f32(false, a, false, bb,
                                                        (short)0, acc, false, false);
        }
#pragma unroll
        for (int v = 0; v < 8; ++v) {
            const int r = v + hi8;
            if (r < 4)
                xr[(size_t)b * (4 * Q_) + (size_t)r * Q_ + q] = acc[v];
        }
    }
}

// ---------------------------------------------------------------------------
// K2: multi-dilation depthwise 3D conv aggregation (tiny; one thread per
// (b,t,h,w), all 4 reduction channels, all 3 convs, weights[] folded in).
// agg[b,r,t,hw] = sum_i weights[i] * (dwconv_i(xr)[r] + b_i[r])
// ---------------------------------------------------------------------------
__global__ void k2_dwconv(const float* __restrict__ xr,
                          const float* __restrict__ Wsa1, const float* __restrict__ bsa1,
                          const float* __restrict__ Wsa2, const float* __restrict__ bsa2,
                          const float* __restrict__ Wsa3, const float* __restrict__ bsa3,
                          const float* __restrict__ wv,
                          float* __restrict__ agg)
{
    __shared__ float Ws[3][4 * 81];
    __shared__ float bs[3][4];
    __shared__ float wvs[3];

    const int tid = threadIdx.x;
    for (int i = tid; i < 324; i += 256) { Ws[0][i] = Wsa1[i]; Ws[1][i] = Wsa2[i]; Ws[2][i] = Wsa3[i]; }
    if (tid < 4) { bs[0][tid] = bsa1[tid]; bs[1][tid] = bsa2[tid]; bs[2][tid] = bsa3[tid]; }
    if (tid < 3) wvs[tid] = wv[tid];
    __syncthreads();

    const int gid = blockIdx.x * 256 + tid;
    if (gid >= B_ * T_ * HW_) return;
    const int b   = gid / (T_ * HW_);
    const int rem = gid % (T_ * HW_);
    const int t   = rem / HW_;
    const int hw  = rem % HW_;
    const int h   = hw / 14, w = hw % 14;

    float out[4] = {0.f, 0.f, 0.f, 0.f};
    for (int ci = 0; ci < 3; ++ci) {
        const int dil = ci + 1;
        for (int r = 0; r < 4; ++r) {
            float s = bs[ci][r];
            const float* xbase = xr + (size_t)(b * 4 + r) * Q_;
            for (int kt = 0; kt < 9; ++kt) {
                const int tt = t + kt - 4;
                if ((unsigned)tt >= (unsigned)T_) continue;
                const float* xt = xbase + (size_t)tt * HW_;
#pragma unroll
                for (int kh = 0; kh < 3; ++kh) {
                    const int hh = h + (kh - 1) * dil;
                    if ((unsigned)hh >= 14u) continue;
#pragma unroll
                    for (int kw = 0; kw < 3; ++kw) {
                        const int ww = w + (kw - 1) * dil;
                        if ((unsigned)ww >= 14u) continue;
                        s += Ws[ci][r * 81 + kt * 9 + kh * 3 + kw] * xt[hh * 14 + ww];
                    }
                }
            }
            out[r] += wvs[ci] * s;
        }
    }
#pragma unroll
    for (int r = 0; r < 4; ++r)
        agg[(size_t)(b * 4 + r) * Q_ + (size_t)t * HW_ + hw] = out[r];
}

// ---------------------------------------------------------------------------
// K3: fused affinity branch, one workgroup per (b,t), 8 waves.
// LDS: X[64][208], Yn[64][208], Yp[64][208], G[16][208], W_back[256]  (~170KB)
// Streaming over sd-blocks of 16:
//   phase A: A_blk = X^T * Y[:,blk]  (13 hw-tiles x 16 wmma k-steps)
//            g = w2 * (sigmoid(A)-0.5), transposed into G
//   phase B: F[c,hw] += Y[:,blk] * G   (52 acc tiles over 8 waves)
// Epilogue: gate = W_back*agg, out = F * (sigmoid(gate)-0.5)
// ---------------------------------------------------------------------------
__global__ void k3_affinity(const float* __restrict__ x,
                            const float* __restrict__ x2,
                            const float* __restrict__ agg,
                            const float* __restrict__ Wb,
                            const float* __restrict__ w2,
                            float* __restrict__ out)
{
    extern __shared__ float smem[];
    float* Xs  = smem;                  // 64*208
    float* Yn  = Xs + 64 * HWP_;        // 64*208
    float* Yp  = Yn + 64 * HWP_;        // 64*208
    float* Gs  = Yp + 64 * HWP_;        // 16*208
    float* Wbs = Gs + 16 * HWP_;        // 256

    const int tid = threadIdx.x;
    const int t   = blockIdx.x;
    const int b   = blockIdx.y;
    const int tn  = (t + 1 < T_) ? t + 1 : T_ - 1;
    const int tp  = (t > 0) ? t - 1 : 0;

    if (tid < 256) Wbs[tid] = Wb[tid];
    for (int i = tid; i < 64 * HWP_; i += 256) {
        const int c = i / HWP_, col = i % HWP_;
        float vx = 0.f, vn = 0.f, vp = 0.f;
        if (col < HW_) {
            const size_t base = (size_t)(b * 64 + c) * Q_;
            vx = x [base + (size_t)t  * HW_ + col];
            vn = x2[base + (size_t)tn * HW_ + col];
            vp = x2[base + (size_t)tp * HW_ + col];
        }
        Xs[i] = vx; Yn[i] = vn; Yp[i] = vp;
    }
    __syncthreads();

    const int lane  = tid & 31;
    const int wave  = tid >> 5;
    const int row   = lane & 15;
    const int hi    = lane >> 4;
    const int khalf = hi * 2;
    const int hi8   = hi * 8;

    v8f facc[7];
#pragma unroll
    for (int i = 0; i < 7; ++i) facc[i] = (v8f){};

    for (int pass = 0; pass < 2; ++pass) {
        const float* Ys = pass ? Yp : Yn;
        const float w2s = w2[pass];
        for (int j = 0; j < 13; ++j) {
            // ---- phase A: affinity tiles for sd-block j ----
            for (int m = wave; m < 13; m += 8) {
                v8f acc = {};
#pragma unroll
                for (int k4 = 0; k4 < 16; ++k4) {
                    const int kk = k4 * 4 + khalf;
                    v2f a, bb;
                    a.x  = Xs[kk * HWP_ + m * 16 + row];        // X^T[hw][c]
                    a.y  = Xs[(kk + 1) * HWP_ + m * 16 + row];
                    bb.x = Ys[kk * HWP_ + j * 16 + row];        // Y[c][sd]
                    bb.y = Ys[(kk + 1) * HWP_ + j * 16 + row];
                    acc = __builtin_amdgcn_wmma_f32_16x16x4_f32(false, a, false, bb,
                                                                (short)0, acc, false, false);
                }
                const int sdg  = j * 16 + row;                  // N = sd column
                const float sc = (sdg < HW_) ? w2s : 0.f;       // zero padded sd
#pragma unroll
                for (int v = 0; v < 8; ++v) {
                    const float g = sc * (1.f / (1.f + __expf(-acc[v])) - 0.5f);
                    Gs[row * HWP_ + m * 16 + v + hi8] = g;      // transposed: G[sd][hw]
                }
            }
            __syncthreads();
            // ---- phase B: accumulate F += Y[:,blk] * G ----
            int idx = 0;
            for (int qv = wave; qv < 52; qv += 8, ++idx) {
                const int mc = qv / 13, nh = qv % 13;
                v8f acc = facc[idx];
#pragma unroll
                for (int k4 = 0; k4 < 4; ++k4) {
                    const int kk = k4 * 4 + khalf;
                    v2f a, bb;
                    a.x  = Ys[(mc * 16 + row) * HWP_ + j * 16 + kk];     // Y[c][sd]
                    a.y  = Ys[(mc * 16 + row) * HWP_ + j * 16 + kk + 1];
                    bb.x = Gs[kk * HWP_ + nh * 16 + row];                // G[sd][hw]
                    bb.y = Gs[(kk + 1) * HWP_ + nh * 16 + row];
                    acc = __builtin_amdgcn_wmma_f32_16x16x4_f32(false, a, false, bb,
                                                                (short)0, acc, false, false);
                }
                facc[idx] = acc;
            }
            __syncthreads();
        }
    }

    // ---- epilogue: gate + writeout ----
    int idx = 0;
    for (int qv = wave; qv < 52; qv += 8, ++idx) {
        const int mc = qv / 13, nh = qv % 13;
        const int hwg = nh * 16 + row;
        if (hwg < HW_) {
            float av[4];
#pragma unroll
            for (int r = 0; r < 4; ++r)
                av[r] = agg[(size_t)(b * 4 + r) * Q_ + (size_t)t * HW_ + hwg];
#pragma unroll
            for (int v = 0; v < 8; ++v) {
                const int c = mc * 16 + v + hi8;
                const float gate = Wbs[c * 4 + 0] * av[0] + Wbs[c * 4 + 1] * av[1]
                                 + Wbs[c * 4 + 2] * av[2] + Wbs[c * 4 + 3] * av[3];
                const float gf = 1.f / (1.f + __expf(-gate)) - 0.5f;
                out[(size_t)(b * 64 + c) * Q_ + (size_t)t * HW_ + hwg] = facc[idx][v] * gf;
            }
        }
    }
}

// ---------------------------------------------------------------------------
extern "C" void kernel_launch(void* const* d_in, const int* in_sizes, int n_in,
                              void* d_out, int out_size, void* d_ws, size_t ws_size,
                              hipStream_t stream)
{
    (void)in_sizes; (void)n_in; (void)out_size; (void)ws_size;
    const float* x       = (const float*)d_in[0];
    const float* W_down  = (const float*)d_in[1];
    const float* W_down2 = (const float*)d_in[2];
    const float* W_sa1   = (const float*)d_in[3];
    const float* b_sa1   = (const float*)d_in[4];
    const float* W_sa2   = (const float*)d_in[5];
    const float* b_sa2   = (const float*)d_in[6];
    const float* W_sa3   = (const float*)d_in[7];
    const float* b_sa3   = (const float*)d_in[8];
    const float* W_back  = (const float*)d_in[9];
    const float* weights = (const float*)d_in[10];
    const float* weights2= (const float*)d_in[11];
    float* out = (float*)d_out;

    float* ws  = (float*)d_ws;
    float* x2  = ws;                               // B*C*Q  = 6,422,528 f
    float* xr  = x2 + (size_t)B_ * C_ * Q_;        // B*R*Q  =   401,408 f
    float* agg = xr + (size_t)B_ * R_ * Q_;        // B*R*Q  =   401,408 f

    // K1: pointwise GEMMs (x2, xr)
    k1_pointwise<<<dim3(Q_ / 128, B_), 256, 0, stream>>>(x, W_down2, W_down, x2, xr);

    // K2: depthwise conv aggregation
    const int n2 = B_ * T_ * HW_;
    k2_dwconv<<<(n2 + 255) / 256, 256, 0, stream>>>(xr, W_sa1, b_sa1, W_sa2, b_sa2,
                                                    W_sa3, b_sa3, weights, agg);

    // K3: fused affinity + gate + writeout (needs >64KB dynamic LDS)
    const size_t smem3 = (size_t)(3 * 64 * HWP_ + 16 * HWP_ + 256) * sizeof(float); // 174,080 B
    static_assert((3 * 64 * HWP_ + 16 * HWP_ + 256) * sizeof(float) <= 320 * 1024, "LDS");
    hipFuncSetAttribute((const void*)k3_affinity,
                        hipFuncAttributeMaxDynamicSharedMemorySize, (int)smem3);
    k3_affinity<<<dim3(T_, B_), 256, smem3, stream>>>(x, x2, agg, W_back, weights2, out);
}